// S4_65532611002457
// MI455X (gfx1250) — compile-verified
//
#include <hip/hip_runtime.h>
#include <math.h>

// ---------------------------------------------------------------------------
// S4D SSM scan, exact chunked-convolution reformulation for MI455X (gfx1250).
//   y[cT+j] = d*u[cT+j] + Re(C * Abar^{j+1} * s_c) + sum_{t<=j} K[t]*u[cT+j-t]
//   s_{c+1} = Abar^T * s_c + v_c,  v_c[n] = sum_j Abar^{T-1-j} Bbar u[cT+j]
// Core compute = one FP32 GEMM Y[256x1024] = A'[256x384] @ B'[384x1024]
// executed with V_WMMA_F32_16X16X4_F32; B' staged to LDS with async copies
// (GLOBAL_LOAD_ASYNC_TO_LDS_B128 + s_wait_asynccnt), double buffered.
// B' is stored K-pair interleaved so a WMMA B-fragment is ONE ds_load_b64.
// ---------------------------------------------------------------------------

#define L_SEQ   262144
#define NST     64
#define T_CHUNK 256
#define NCH     (L_SEQ / T_CHUNK)      // 1024 chunks
#define KTOT    (T_CHUNK + 2 * NST)    // 384 = conv rows + Re/Im state rows

// Workspace layout (float offsets). Total ~3.2 MB.
#define OFF_K   0                              // K[256]
#define OFF_PT  (OFF_K  + T_CHUNK)             // Pt[t][n][2]  (chunk-reduce weights)
#define OFF_WT  (OFF_PT + T_CHUNK * NST * 2)   // Wt[t][n][2]  (boundary weights C*Abar^{t+1})
#define OFF_AT  (OFF_WT + T_CHUNK * NST * 2)   // Abar^T [n][2]
#define OFF_V   (OFF_AT + NST * 2)             // v[c][n][2]
#define OFF_S   (OFF_V  + NCH * NST * 2)       // s[c][n][2]  (exclusive prefix states)
#define OFF_AP  (OFF_S  + NCH * NST * 2)       // A' [256][384]
#define OFF_BP  (OFF_AP + T_CHUNK * KTOT)      // B' [192 kpairs][1024 cols][2]

typedef __attribute__((ext_vector_type(2))) float v2f;
typedef __attribute__((ext_vector_type(4))) float v4f;
typedef __attribute__((ext_vector_type(8))) float v8f;
typedef int v4i __attribute__((vector_size(16)));
typedef __attribute__((address_space(1))) v4i* global_v4i_ptr;   // HIP: __device__
typedef __attribute__((address_space(3))) v4i* lds_v4i_ptr;      // HIP: __shared__

#if defined(__has_builtin)
#if __has_builtin(__builtin_amdgcn_global_load_async_to_lds_b128)
#define HAVE_ASYNC_LDS 1
#endif
#endif

// ---------------------------------------------------------------------------
// Kernel 1: discretization + kernel/weight tables. One block, 64 threads.
// Deterministic LDS reduction for K (no float atomics -> bitwise reproducible).
// ---------------------------------------------------------------------------
__global__ __launch_bounds__(64) void s4_setup(const float* __restrict__ lam_r,
                                               const float* __restrict__ lam_i,
                                               const float* __restrict__ bvec,
                                               const float* __restrict__ cvec,
                                               const float* __restrict__ delta,
                                               float* __restrict__ ws) {
  __shared__ float contrib[128][NST];          // 32 KB, half of K at a time
  const int n = threadIdx.x;                   // state index 0..63

  const float step = expf(delta[0]);
  const float lr = fminf(lam_r[n], -1e-4f);    // clip: strictly stable real part
  const float li = lam_i[n];

  // Abar = exp(step*lam)   (complex exponential)
  const float er  = expf(step * lr);
  const float Are = er * cosf(step * li);
  const float Aim = er * sinf(step * li);

  // Bbar = (Abar - 1)/lam * b
  const float inv_d2 = 1.0f / (lr * lr + li * li);
  const float numr = Are - 1.0f, numi = Aim;
  const float Bre = (numr * lr + numi * li) * inv_d2 * bvec[n];
  const float Bim = (numi * lr - numr * li) * inv_d2 * bvec[n];

  const float Cre = cvec[2 * n], Cim = cvec[2 * n + 1];
  const float CBr = Cre * Bre - Cim * Bim;     // C*Bbar
  const float CBi = Cre * Bim + Cim * Bre;

  float* Pt = ws + OFF_PT;
  float* Wt = ws + OFF_WT;

  float pwr = 1.0f, pwi = 0.0f;                // Abar^t, t starts at 0
  for (int half = 0; half < 2; ++half) {
    for (int tl = 0; tl < 128; ++tl) {
      const int t = half * 128 + tl;
      // K[t] contribution: Re(C*Bbar*Abar^t)
      contrib[tl][n] = CBr * pwr - CBi * pwi;
      // P[n][T-1-t] = Abar^t * Bbar  (stored transposed: Pt[t'][n])
      const int tp = T_CHUNK - 1 - t;
      Pt[tp * 2 * NST + 2 * n]     = pwr * Bre - pwi * Bim;
      Pt[tp * 2 * NST + 2 * n + 1] = pwr * Bim + pwi * Bre;
      // advance power: pw = pw * Abar  -> now Abar^{t+1}
      const float nr = pwr * Are - pwi * Aim;
      const float ni = pwr * Aim + pwi * Are;
      pwr = nr; pwi = ni;
      // W[n][t] = C * Abar^{t+1}  (stored transposed: Wt[t][n])
      Wt[t * 2 * NST + 2 * n]     = Cre * pwr - Cim * pwi;
      Wt[t * 2 * NST + 2 * n + 1] = Cre * pwi + Cim * pwr;
    }
    __syncthreads();
    // ordered (deterministic) reduction over n for this half's 128 t values
    for (int q = 0; q < 2; ++q) {
      const int tl = n * 2 + q;
      float s = 0.0f;
      for (int m = 0; m < NST; ++m) s += contrib[tl][m];
      ws[OFF_K + half * 128 + tl] = s;
    }
    __syncthreads();
  }
  // Abar^T (chunk transition)
  ws[OFF_AT + 2 * n]     = pwr;
  ws[OFF_AT + 2 * n + 1] = pwi;
}

// ---------------------------------------------------------------------------
// Kernel 2: materialize A' = [ LowerToeplitz(K) | Re(W)^T | -Im(W)^T ]  (256x384)
// ---------------------------------------------------------------------------
__global__ void s4_build_a(float* __restrict__ ws) {
  const int idx = blockIdx.x * blockDim.x + threadIdx.x;
  if (idx >= T_CHUNK * KTOT) return;
  const int j = idx / KTOT;                    // output row (within-chunk index)
  const int k = idx % KTOT;                    // reduction index
  float val;
  if (k < T_CHUNK) {
    val = (k <= j) ? ws[OFF_K + (j - k)] : 0.0f;
  } else if (k < T_CHUNK + NST) {
    const int n = k - T_CHUNK;
    val = ws[OFF_WT + j * 2 * NST + 2 * n];            // Re(W[n][j])
  } else {
    const int n = k - T_CHUNK - NST;
    val = -ws[OFF_WT + j * 2 * NST + 2 * n + 1];       // -Im(W[n][j])
  }
  ws[OFF_AP + idx] = val;
}

// ---------------------------------------------------------------------------
// Kernel 3: per-chunk input reduction  v_c[n] = sum_j Pt[j][n] * u[cT+j]
// (the parallelized bulk of the scan's flops; 1024 independent blocks)
// ---------------------------------------------------------------------------
__global__ __launch_bounds__(256) void s4_chunk_reduce(const float* __restrict__ u,
                                                       float* __restrict__ ws) {
  __shared__ float ush[T_CHUNK];
  __shared__ float part[4][NST][2];
  const int c = blockIdx.x;
  const int tid = threadIdx.x;
  ush[tid] = u[c * T_CHUNK + tid];
  __syncthreads();

  const int n = tid & 63, seg = tid >> 6;      // 4 segments of 64 timesteps
  const float* Pt = ws + OFF_PT;
  float ar = 0.0f, ai = 0.0f;
  for (int jj = 0; jj < 64; ++jj) {
    const int j = seg * 64 + jj;
    const float uv = ush[j];
    ar += Pt[j * 2 * NST + 2 * n] * uv;        // coalesced across n
    ai += Pt[j * 2 * NST + 2 * n + 1] * uv;
  }
  part[seg][n][0] = ar; part[seg][n][1] = ai;
  __syncthreads();
  if (tid < NST) {
    const float vr = part[0][tid][0] + part[1][tid][0] + part[2][tid][0] + part[3][tid][0];
    const float vi = part[0][tid][1] + part[1][tid][1] + part[2][tid][1] + part[3][tid][1];
    ws[OFF_V + c * 2 * NST + 2 * tid]     = vr;
    ws[OFF_V + c * 2 * NST + 2 * tid + 1] = vi;
  }
}

// ---------------------------------------------------------------------------
// Kernel 4: exclusive prefix over 1024 chunk states (per-lane independent).
// ---------------------------------------------------------------------------
__global__ __launch_bounds__(64) void s4_scan(float* __restrict__ ws) {
  const int n = threadIdx.x;
  const float atr = ws[OFF_AT + 2 * n], ati = ws[OFF_AT + 2 * n + 1];
  float sr = 0.0f, si = 0.0f;
  for (int c = 0; c < NCH; ++c) {
    ws[OFF_S + c * 2 * NST + 2 * n]     = sr;  // state ENTERING chunk c
    ws[OFF_S + c * 2 * NST + 2 * n + 1] = si;
    const float vr = ws[OFF_V + c * 2 * NST + 2 * n];
    const float vi = ws[OFF_V + c * 2 * NST + 2 * n + 1];
    const float nr = atr * sr - ati * si + vr;
    const float ni = atr * si + ati * sr + vi;
    sr = nr; si = ni;
  }
}

// ---------------------------------------------------------------------------
// Kernel 5: materialize B' = [ U(chunk-major) ; Re(S) ; Im(S) ], stored
// K-PAIR INTERLEAVED: element (k,c) -> (k>>1)*2*NCH + 2*c + (k&1), so the
// WMMA B-fragment rows {kl, kl+1} (kl even) are 8 contiguous bytes.
// ---------------------------------------------------------------------------
__global__ void s4_build_b(const float* __restrict__ u, float* __restrict__ ws) {
  const int idx = blockIdx.x * blockDim.x + threadIdx.x;
  if (idx >= KTOT * NCH) return;
  const int k = idx / NCH;
  const int c = idx % NCH;
  float val;
  if (k < T_CHUNK)            val = u[c * T_CHUNK + k];
  else if (k < T_CHUNK + NST) val = ws[OFF_S + c * 2 * NST + 2 * (k - T_CHUNK)];
  else                        val = ws[OFF_S + c * 2 * NST + 2 * (k - T_CHUNK - NST) + 1];
  ws[OFF_BP + (k >> 1) * 2 * NCH + 2 * c + (k & 1)] = val;
}

// ---------------------------------------------------------------------------
// Kernel 6: Y = A' @ B' via V_WMMA_F32_16X16X4_F32, fused epilogue y += d*u.
// Block = 128 threads (4 wave32s); each wave owns a 16x64 output strip.
// The shared 64(K)x64(N) B' strip is staged to LDS once per block per phase
// (async global->LDS b128 copies, double buffered, s_wait_asynccnt handoff).
// LDS kpair-row stride 160 floats: adjacent kpair rows (read by the two
// half-waves of a fragment ds_load_b64) sit 32 banks apart -> conflict-free.
// ---------------------------------------------------------------------------
#define KB    64                // K-rows per staging phase
#define KPH   (KB / 2)          // 32 kpair-rows per phase
#define NPH   (KTOT / KB)       // 6 phases
#define BSTR2 160               // padded LDS kpair-row stride (floats)

__device__ __forceinline__ void s4_wait_async() {
#if defined(HAVE_ASYNC_LDS)
#if __has_builtin(__builtin_amdgcn_s_wait_asynccnt)
  __builtin_amdgcn_s_wait_asynccnt(0);
#else
  asm volatile("s_wait_asynccnt 0" ::: "memory");
#endif
#endif
}

// Stage phase p of B' (column block n0, 32 kpair-rows x 128 floats) into LDS.
__device__ __forceinline__ void s4_stage_b(const float* __restrict__ Bp, int p,
                                           float* __restrict__ dst, int n0, int tid) {
#pragma unroll
  for (int q = 0; q < 8; ++q) {
    const int chunk = q * 128 + tid;           // 1024 x 16-byte chunks
    const int kp = chunk >> 5;                 // 0..31  (kpair row within phase)
    const int co = (chunk & 31) << 2;          // 0..124 (float offset within row)
    const float* g = Bp + (size_t)(p * KPH + kp) * (2 * NCH) + 2 * n0 + co;
    float* l = dst + kp * BSTR2 + co;
#if defined(HAVE_ASYNC_LDS)
    __builtin_amdgcn_global_load_async_to_lds_b128(
        (global_v4i_ptr)const_cast<float*>(g), (lds_v4i_ptr)l, 0, 0);
#else
    *(v4f*)l = *(const v4f*)g;
#endif
  }
}

__global__ __launch_bounds__(128) void s4_gemm(const float* __restrict__ u,
                                               const float* __restrict__ dptr,
                                               const float* __restrict__ ws,
                                               float* __restrict__ out) {
  __shared__ __align__(16) float Bsh[2][KPH * BSTR2];  // 2 x 20 KB, double buffered
  const float* Ap = ws + OFF_AP;
  const float* Bp = ws + OFF_BP;
  const int tid  = threadIdx.x;
  const int wave = tid >> 5;
  const int lane = tid & 31;
  const int l15 = lane & 15, hi = lane >> 4;
  const int m0 = blockIdx.y * 64 + wave * 16;  // output rows (within-chunk j)
  const int n0 = blockIdx.x * 64;              // output cols (chunk index c)

  s4_stage_b(Bp, 0, Bsh[0], n0, tid);          // prime the pipeline
  s4_wait_async();
  __syncthreads();

  v8f acc[4] = {};
  const float* arow = Ap + (m0 + l15) * KTOT;

  for (int p = 0; p < NPH; ++p) {
    if (p + 1 < NPH) s4_stage_b(Bp, p + 1, Bsh[(p + 1) & 1], n0, tid);
    const float* bs = Bsh[p & 1];
    __builtin_prefetch(arow + (p + 1) * KB, 0, 0);     // global_prefetch_b8

    for (int kb = 0; kb < KB; kb += 4) {
      const int kl = kb + (hi << 1);                   // fragment K rows kl, kl+1
      const v2f a = *(const v2f*)(arow + p * KB + kl); // A[m0+l15][..] (b64 load)
      const int kp = (kb >> 1) + hi;                   // local kpair row
#pragma unroll
      for (int t = 0; t < 4; ++t) {
        const int col = t * 16 + l15;
        // rows {kl, kl+1} at column col: contiguous pair -> single ds_load_b64
        const v2f b = *(const v2f*)(bs + kp * BSTR2 + 2 * col);
        acc[t] = __builtin_amdgcn_wmma_f32_16x16x4_f32(
            /*neg_a=*/false, a, /*neg_b=*/false, b,
            /*c_mod=*/(short)0, acc[t], /*reuse_a=*/false, /*reuse_b=*/false);
      }
    }
    s4_wait_async();   // staged phase p+1 fully in LDS
    __syncthreads();   // and everyone done reading phase p's buffer
  }

  // Epilogue: lane's acc[t][0..7] covers 8 consecutive rows -> b128 traffic.
  const float dv = dptr[0];
#pragma unroll
  for (int t = 0; t < 4; ++t) {
    const int col = n0 + t * 16 + l15;                 // chunk c
    const int gbase = col * T_CHUNK + m0 + hi * 8;     // 8 consecutive timesteps
    const v4f u0 = *(const v4f*)(u + gbase);
    const v4f u1 = *(const v4f*)(u + gbase + 4);
    const v4f a0 = {acc[t][0], acc[t][1], acc[t][2], acc[t][3]};
    const v4f a1 = {acc[t][4], acc[t][5], acc[t][6], acc[t][7]};
    *(v4f*)(out + gbase)     = a0 + dv * u0;
    *(v4f*)(out + gbase + 4) = a1 + dv * u1;
  }
}

// ---------------------------------------------------------------------------
extern "C" void kernel_launch(void* const* d_in, const int* in_sizes, int n_in,
                              void* d_out, int out_size, void* d_ws, size_t ws_size,
                              hipStream_t stream) {
  (void)in_sizes; (void)n_in; (void)out_size; (void)ws_size;
  const float* u     = (const float*)d_in[0];
  const float* lam_r = (const float*)d_in[1];
  const float* lam_i = (const float*)d_in[2];
  const float* bvec  = (const float*)d_in[3];
  const float* cvec  = (const float*)d_in[4];
  const float* dvec  = (const float*)d_in[5];
  const float* delta = (const float*)d_in[6];
  float* ws  = (float*)d_ws;   // needs ~3.2 MB
  float* out = (float*)d_out;

  s4_setup<<<1, 64, 0, stream>>>(lam_r, lam_i, bvec, cvec, delta, ws);
  s4_build_a<<<(T_CHUNK * KTOT + 255) / 256, 256, 0, stream>>>(ws);
  s4_chunk_reduce<<<NCH, 256, 0, stream>>>(u, ws);
  s4_scan<<<1, 64, 0, stream>>>(ws);
  s4_build_b<<<(KTOT * NCH + 255) / 256, 256, 0, stream>>>(u, ws);
  s4_gemm<<<dim3(NCH / 64, T_CHUNK / 64), 128, 0, stream>>>(u, dvec, ws, out);
}